// Model_57011395887417
// MI455X (gfx1250) — compile-verified
//
#include <hip/hip_runtime.h>
#include <math.h>

#define NN 207
#define NP 208
#define DH 64
#define BB 64
#define TT 12
#define ED 10
#define HH 12
#define AT_STRIDE 210   // LDS row stride for A tiles (conflict-free)

typedef float v2f __attribute__((ext_vector_type(2)));
typedef float v4f __attribute__((ext_vector_type(4)));
typedef float v8f __attribute__((ext_vector_type(8)));

// ---- fp32 WMMA 16x16x4: D = A(16x4) * B(4x16) + C ----
// A frag: lane L holds A[m=L%16][kbase+0..1], kbase=(L/16)*2
// B frag: lane L holds B[kbase+0..1][n=L%16]
// C/D:    lane L vgpr i holds D[m=i+(L/16)*8][n=L%16]
__device__ __forceinline__ v8f wmma4(float a0, float a1, float b0, float b1, v8f c) {
  v2f a; a.x = a0; a.y = a1;
  v2f b; b.x = b0; b.y = b1;
  return __builtin_amdgcn_wmma_f32_16x16x4_f32(false, a, false, b, (short)0, c, false, false);
}

// ---------------- utility ----------------
__global__ void k_zero(float* __restrict__ p, int n) {
  int i = blockIdx.x * blockDim.x + threadIdx.x;
  int stride = gridDim.x * blockDim.x;
  for (; i < n; i += stride) p[i] = 0.f;
}

// ---------------- adjacency front-end ----------------
__global__ void k_conv1(const float* __restrict__ xf, const float* __restrict__ w,
                        const float* __restrict__ b, float* __restrict__ out) {
  int idx = blockIdx.x * blockDim.x + threadIdx.x;
  if (idx >= NN * 8 * 91) return;
  int t = idx % 91, o = (idx / 91) % 8, n = idx / (91 * 8);
  float acc = b[o];
#pragma unroll
  for (int j = 0; j < 10; ++j) acc += xf[n * 100 + t + j] * w[o * 10 + j];
  out[(n * 8 + o) * 91 + t] = fmaxf(acc, 0.f);
}

__global__ void k_stats_conv(const float* __restrict__ buf, int C, int L,
                             float* __restrict__ mean, float* __restrict__ rstd) {
  int ch = blockIdx.x, tid = threadIdx.x;
  int cnt = NN * L;
  float s = 0.f, q = 0.f;
  for (int i = tid; i < cnt; i += 256) {
    int n = i / L, t = i - n * L;
    float v = buf[(n * C + ch) * L + t];
    s += v; q += v * v;
  }
  __shared__ float ss[256], sq[256];
  ss[tid] = s; sq[tid] = q; __syncthreads();
  for (int w = 128; w > 0; w >>= 1) {
    if (tid < w) { ss[tid] += ss[tid + w]; sq[tid] += sq[tid + w]; }
    __syncthreads();
  }
  if (tid == 0) {
    float m = ss[0] / cnt;
    float var = sq[0] / cnt - m * m;
    mean[ch] = m;
    rstd[ch] = rsqrtf(var + 1e-5f);
  }
}

__global__ void k_conv2(const float* __restrict__ b1, const float* __restrict__ m1,
                        const float* __restrict__ r1, const float* __restrict__ w,
                        const float* __restrict__ b, float* __restrict__ out) {
  int idx = blockIdx.x * blockDim.x + threadIdx.x;
  if (idx >= NN * 16 * 82) return;
  int t = idx % 82, o = (idx / 82) % 16, n = idx / (82 * 16);
  float acc = b[o];
  for (int i = 0; i < 8; ++i) {
    float mm = m1[i], rr = r1[i];
#pragma unroll
    for (int j = 0; j < 10; ++j) {
      float v = (b1[(n * 8 + i) * 91 + t + j] - mm) * rr;
      acc += v * w[(o * 8 + i) * 10 + j];
    }
  }
  out[(n * 16 + o) * 82 + t] = fmaxf(acc, 0.f);
}

__global__ void k_fc(const float* __restrict__ b2, const float* __restrict__ m2,
                     const float* __restrict__ r2, const float* __restrict__ w,
                     const float* __restrict__ b, float* __restrict__ fx) {
  int idx = blockIdx.x * blockDim.x + threadIdx.x;
  if (idx >= NN * ED) return;
  int f = idx % ED, n = idx / ED;
  float acc = b[f];
  for (int ch = 0; ch < 16; ++ch) {
    float mm = m2[ch], rr = r2[ch];
    for (int t = 0; t < 82; ++t) {
      float v = (b2[(n * 16 + ch) * 82 + t] - mm) * rr;
      acc += v * w[f * 1312 + ch * 82 + t];
    }
  }
  fx[n * ED + f] = fmaxf(acc, 0.f);
}

__global__ void k_stats_feat(const float* __restrict__ buf, int C,
                             float* __restrict__ mean, float* __restrict__ rstd) {
  int ch = blockIdx.x, tid = threadIdx.x;
  float s = 0.f, q = 0.f;
  for (int n = tid; n < NN; n += 256) {
    float v = buf[n * C + ch];
    s += v; q += v * v;
  }
  __shared__ float ss[256], sq[256];
  ss[tid] = s; sq[tid] = q; __syncthreads();
  for (int w = 128; w > 0; w >>= 1) {
    if (tid < w) { ss[tid] += ss[tid + w]; sq[tid] += sq[tid + w]; }
    __syncthreads();
  }
  if (tid == 0) {
    float m = ss[0] / NN;
    float var = sq[0] / NN - m * m;
    mean[ch] = m;
    rstd[ch] = rsqrtf(var + 1e-5f);
  }
}

__global__ void k_pairs(const float* __restrict__ fx, const float* __restrict__ mf,
                        const float* __restrict__ rf, const float* __restrict__ fow,
                        const float* __restrict__ fob, const float* __restrict__ fcw,
                        const float* __restrict__ fcb, const float* __restrict__ U,
                        float* __restrict__ adj) {
  int p = blockIdx.x * blockDim.x + threadIdx.x;
  if (p >= NN * NN) return;
  int i = p / NN, j = p - i * NN;
  float xs[ED], xr[ED];
#pragma unroll
  for (int f = 0; f < ED; ++f) {
    xs[f] = (fx[j * ED + f] - mf[f]) * rf[f];  // send = x[j]
    xr[f] = (fx[i * ED + f] - mf[f]) * rf[f];  // recv = x[i]
  }
  float lg0 = fcb[0], lg1 = fcb[1];
  for (int f = 0; f < ED; ++f) {
    float acc = fob[f];
#pragma unroll
    for (int q = 0; q < ED; ++q) acc += fow[f * 20 + q] * xs[q] + fow[f * 20 + ED + q] * xr[q];
    acc = fmaxf(acc, 0.f);
    lg0 += fcw[f] * acc;
    lg1 += fcw[ED + f] * acc;
  }
  float u0 = U[p * 2], u1 = U[p * 2 + 1];
  float g0 = -logf(-logf(u0 + 1e-20f) + 1e-20f);
  float g1 = -logf(-logf(u1 + 1e-20f) + 1e-20f);
  // argmax of softmax((logits+g)/TEMP) == argmax of (logits+g); ties -> index 0
  adj[p] = ((lg0 + g0) >= (lg1 + g1)) ? 1.f : 0.f;
}

__global__ void k_arow(const float* __restrict__ E, const float* __restrict__ adj,
                       float* __restrict__ Apad) {
  __shared__ float vals[NN];
  __shared__ float red[256], red2[256];
  int i = blockIdx.x, tid = threadIdx.x;
  float ei[ED];
#pragma unroll
  for (int e = 0; e < ED; ++e) ei[e] = E[i * ED + e];
  float mx = -1e30f, rs = 0.f;
  for (int j = tid; j < NN; j += 256) {
    float d = 0.f;
#pragma unroll
    for (int e = 0; e < ED; ++e) d += ei[e] * E[j * ED + e];
    d = fmaxf(d, 0.f);
    vals[j] = d;
    mx = fmaxf(mx, d);
    rs += adj[i * NN + j];
  }
  red[tid] = mx; red2[tid] = rs; __syncthreads();
  for (int w = 128; w > 0; w >>= 1) {
    if (tid < w) { red[tid] = fmaxf(red[tid], red[tid + w]); red2[tid] += red2[tid + w]; }
    __syncthreads();
  }
  float MX = red[0], RS = red2[0];
  __syncthreads();
  float se = 0.f;
  for (int j = tid; j < NN; j += 256) se += expf(vals[j] - MX);
  red[tid] = se; __syncthreads();
  for (int w = 128; w > 0; w >>= 1) {
    if (tid < w) red[tid] += red[tid + w];
    __syncthreads();
  }
  float inv = 1.f / red[0];
  float rinv = 1.f / (RS + 1e-8f);
  for (int j = tid; j < NN; j += 256)
    Apad[i * NP + j] = 0.5f * (expf(vals[j] - MX) * inv + adj[i * NN + j] * rinv);
}

// ---------------- temporal normalization ----------------
__global__ void k_norm(const float* __restrict__ xe, float* __restrict__ S,
                       float* __restrict__ MEANB, float* __restrict__ STDB) {
  int idx = blockIdx.x * blockDim.x + threadIdx.x;
  if (idx >= BB * NN) return;
  int n = idx % NN, b = idx / NN;
  float v[TT], s = 0.f;
#pragma unroll
  for (int t = 0; t < TT; ++t) { v[t] = xe[(b * TT + t) * NN + n]; s += v[t]; }
  float m = s / TT, q = 0.f;
#pragma unroll
  for (int t = 0; t < TT; ++t) { float d = v[t] - m; q += d * d; }
  float sd = sqrtf(q / TT + 1e-5f);
  float inv = 1.f / sd;
#pragma unroll
  for (int t = 0; t < TT; ++t) S[(t * NN + n) * BB + b] = (v[t] - m) * inv;
  MEANB[n * BB + b] = m;
  STDB[n * BB + b] = sd;
}

// ---------------- node-adapted weights: W[n] = sum_e E[n,e] * Wpool[e] ----------------
__global__ void k_wn(const float* __restrict__ E, const float* __restrict__ Wp,
                     float* __restrict__ out, int KD, int No, int total) {
  int idx = blockIdx.x * blockDim.x + threadIdx.x;
  if (idx >= total) return;
  int o = idx % No;
  int kd = (idx / No) % KD;
  int n = idx / (No * KD);
  float acc = 0.f;
#pragma unroll
  for (int e = 0; e < ED; ++e) acc += E[n * ED + e] * Wp[(e * KD + kd) * No + o];
  out[idx] = acc;  // layout [n][kd][o]
}

__global__ void k_bias(const float* __restrict__ E, const float* __restrict__ bp,
                       float* __restrict__ out, int No) {
  int idx = blockIdx.x * blockDim.x + threadIdx.x;
  if (idx >= NN * No) return;
  int o = idx % No, n = idx / No;
  float acc = 0.f;
#pragma unroll
  for (int e = 0; e < ED; ++e) acc += E[n * ED + e] * bp[e * No + o];
  out[idx] = acc;
}

// ---------------- RNN: build concatenated input R[m][(c)*64+b] ----------------
__global__ void k_build(const float* __restrict__ X, int dix,
                        const float* __restrict__ H, const float* __restrict__ ZR,
                        int isCand, float* __restrict__ R, int di, int total) {
  int idx = blockIdx.x * blockDim.x + threadIdx.x;
  if (idx >= total) return;
  int b = idx & 63;
  int c = (idx >> 6) % di;
  int m = (idx >> 6) / di;
  float v;
  if (c < dix) {
    v = X[(m * dix + c) * 64 + b];
  } else {
    int ch = c - dix;
    float h = H[(m * DH + ch) * 64 + b];
    if (isCand) h *= ZR[(m * 128 + DH + ch) * 64 + b];  // r-gate * h
    v = h;
  }
  R[idx] = v;  // idx == (m*di + c)*64 + b
}

// ---------------- supports GEMM: XG1[n][c*64+b] = sum_m Apad[n,m] * R[m][c*64+b] ----------------
// Output buffer is padded to 208 rows -> unconditional stores (no exec-mask epilogue).
__global__ void __launch_bounds__(128)
k_supports(const float* __restrict__ Apad, const float* __restrict__ R,
           float* __restrict__ Out, int cols) {
  __shared__ float As[16 * AT_STRIDE];
  int tid = threadIdx.x;
  int blockRow = blockIdx.x;
  for (int idx = tid; idx < 16 * NP; idx += 128) {
    int r = idx / NP, c = idx - r * NP;
    As[r * AT_STRIDE + c] = Apad[(blockRow * 16 + r) * NP + c];
  }
  __syncthreads();
  int wv = tid >> 5, lane = tid & 31;
  int ln = lane & 15, kg = lane >> 4;
  int col = blockIdx.y * 64 + wv * 16 + ln;
  const float* Arow = &As[ln * AT_STRIDE];
  const float* Rc = R + col;
  v8f acc;
#pragma unroll
  for (int i = 0; i < 8; ++i) acc[i] = 0.f;
#pragma unroll 4
  for (int kk = 0; kk < NP; kk += 4) {
    int kb = kk + kg * 2;
    // prefetch the B panel 16 rows (4 k-steps) ahead (speculative; OOB is dropped)
    __builtin_prefetch(Rc + (kb + 16) * cols, 0, 1);
    float a0 = Arow[kb];
    float a1 = Arow[kb + 1];
    float b0 = Rc[kb * cols];
    float b1 = Rc[(kb + 1) * cols];
    acc = wmma4(a0, a1, b0, b1, acc);
  }
  float* dst = Out + (blockRow * 16 + kg * 8) * cols + col;
#pragma unroll
  for (int i = 0; i < 8; ++i) dst[i * cols] = acc[i];
}

// ---------------- per-node GEMM: Out[n][o][b] = act( XG[n] @ W[n] + bias[n] ) ----------------
// XG rows: [0,di) from R (k=0 identity support), [di,2di) from XG1 (k=1, A@input)
template <int ACT>  // 0 = sigmoid, 1 = tanh
__global__ void __launch_bounds__(128)
k_gconv(const float* __restrict__ R, const float* __restrict__ XG1,
        const float* __restrict__ Wn, const float* __restrict__ Bn,
        float* __restrict__ Out, int di, int No, int KC) {
  __shared__ float XGs[16 * 64];
  __shared__ float Ws[16 * 16];
  int n = blockIdx.x, ct = blockIdx.y;
  int tid = threadIdx.x, wv = tid >> 5, lane = tid & 31;
  int ln = lane & 15, kg = lane >> 4;
  int KD = 2 * di;
  const float* Rn = R + n * di * 64;
  const float* Xn = XG1 + n * di * 64;
  const float* Wnn = Wn + n * KD * No;
  float bias = Bn[n * No + ct * 16 + ln];
  v8f acc;
#pragma unroll
  for (int i = 0; i < 8; ++i) acc[i] = 0.f;
  for (int c0 = 0; c0 < KC; ++c0) {
    int kc = c0 * 16;
    for (int idx = tid; idx < 16 * 64; idx += 128) {
      int r = idx >> 6, m = idx & 63;
      int g = kc + r;
      float v = 0.f;
      if (g < di) v = Rn[g * 64 + m];
      else if (g < KD) v = Xn[(g - di) * 64 + m];
      XGs[idx] = v;
    }
    for (int idx = tid; idx < 256; idx += 128) {
      int r = idx >> 4, c = idx & 15;
      int g = kc + r;
      Ws[idx] = (g < KD) ? Wnn[g * No + ct * 16 + c] : 0.f;
      // prefetch next chunk of the node's weight panel
      __builtin_prefetch(Wnn + (g + 16) * No + ct * 16 + c, 0, 1);
    }
    __syncthreads();
#pragma unroll
    for (int s = 0; s < 4; ++s) {
      int kb = s * 4 + kg * 2;
      float a0 = XGs[kb * 64 + wv * 16 + ln];
      float a1 = XGs[(kb + 1) * 64 + wv * 16 + ln];
      float b0 = Ws[kb * 16 + ln];
      float b1 = Ws[(kb + 1) * 16 + ln];
      acc = wmma4(a0, a1, b0, b1, acc);
    }
    __syncthreads();
  }
  float res[8];
#pragma unroll
  for (int i = 0; i < 8; ++i) {
    float v = acc[i] + bias;
    res[i] = (ACT == 0) ? (1.f / (1.f + expf(-v))) : tanhf(v);
  }
  float* dst = Out + (n * No + ct * 16 + ln) * 64 + wv * 16 + kg * 8;
  v4f r0, r1;
#pragma unroll
  for (int i = 0; i < 4; ++i) { r0[i] = res[i]; r1[i] = res[4 + i]; }
  *(v4f*)dst = r0;
  *((v4f*)dst + 1) = r1;
}

// ---------------- GRU-style state update ----------------
__global__ void k_update(const float* __restrict__ ZR, const float* __restrict__ HC,
                         float* __restrict__ H, float* __restrict__ cur, int writeCur) {
  int idx = blockIdx.x * blockDim.x + threadIdx.x;
  if (idx >= NN * DH * BB) return;
  int b = idx & 63;
  int ch = (idx >> 6) & 63;
  int n = idx >> 12;
  float z = ZR[(n * 128 + ch) * 64 + b];
  float h = H[idx];
  float hc = HC[idx];
  float hn = z * h + (1.f - z) * hc;
  H[idx] = hn;
  if (writeCur) cur[idx] = hn;
}

// ---------------- output head ----------------
__global__ void k_out(const float* __restrict__ H, const float* __restrict__ ECW,
                      const float* __restrict__ ECB, const float* __restrict__ EPS,
                      const float* __restrict__ STD, const float* __restrict__ MEAN,
                      float* __restrict__ out) {
  int idx = blockIdx.x * blockDim.x + threadIdx.x;
  if (idx >= BB * HH * NN) return;
  int n = idx % NN;
  int h = (idx / NN) % HH;
  int b = idx / (NN * HH);
  float mu = ECB[2 * h], lv = ECB[2 * h + 1];
  for (int d = 0; d < DH; ++d) {
    float hv = H[(n * DH + d) * 64 + b];
    mu += ECW[(2 * h) * DH + d] * hv;
    lv += ECW[(2 * h + 1) * DH + d] * hv;
  }
  float eps = EPS[idx];
  float o = mu + eps * expf(0.5f * lv);
  o = o * STD[n * 64 + b] + MEAN[n * 64 + b];
  const int BHN = BB * HH * NN;
  out[idx] = o;
  out[BHN + idx] = mu;
  out[2 * BHN + idx] = lv;
}

static inline int cdiv(int a, int b) { return (a + b - 1) / b; }

extern "C" void kernel_launch(void* const* d_in, const int* in_sizes, int n_in,
                              void* d_out, int out_size, void* d_ws, size_t ws_size,
                              hipStream_t stream) {
  (void)in_sizes; (void)out_size; (void)ws_size;
  const float* Xe  = (const float*)d_in[0];
  const float* NF  = (const float*)d_in[1];
  const float* C1W = (const float*)d_in[2];
  const float* C1B = (const float*)d_in[3];
  const float* C2W = (const float*)d_in[4];
  const float* C2B = (const float*)d_in[5];
  const float* FCW = (const float*)d_in[6];
  const float* FCB = (const float*)d_in[7];
  const float* FOW = (const float*)d_in[8];
  const float* FOB = (const float*)d_in[9];
  const float* FC2W = (const float*)d_in[10];
  const float* FC2B = (const float*)d_in[11];
  const float* E   = (const float*)d_in[12];
  const float *Wg[2], *Bg[2], *Wc[2], *Bc[2], *U, *EPS, *ECW, *ECB;
  if (n_in >= 25) {  // lists expanded per-element
    Wg[0] = (const float*)d_in[13]; Wg[1] = (const float*)d_in[14];
    Bg[0] = (const float*)d_in[15]; Bg[1] = (const float*)d_in[16];
    Wc[0] = (const float*)d_in[17]; Wc[1] = (const float*)d_in[18];
    Bc[0] = (const float*)d_in[19]; Bc[1] = (const float*)d_in[20];
    U = (const float*)d_in[21]; EPS = (const float*)d_in[22];
    ECW = (const float*)d_in[23]; ECB = (const float*)d_in[24];
  } else {           // lists concatenated flat
    const float* wgc = (const float*)d_in[13]; Wg[0] = wgc; Wg[1] = wgc + 10 * 2 * 65 * 128;
    const float* bgc = (const float*)d_in[14]; Bg[0] = bgc; Bg[1] = bgc + 10 * 128;
    const float* wcc = (const float*)d_in[15]; Wc[0] = wcc; Wc[1] = wcc + 10 * 2 * 65 * 64;
    const float* bcc = (const float*)d_in[16]; Bc[0] = bcc; Bc[1] = bcc + 10 * 64;
    U = (const float*)d_in[17]; EPS = (const float*)d_in[18];
    ECW = (const float*)d_in[19]; ECB = (const float*)d_in[20];
  }

  // ---- workspace carve-out (floats, 64-float aligned) ----
  float* ws = (float*)d_ws;
  size_t off = 0;
  auto alloc = [&](size_t nf) { float* p = ws + off; off += (nf + 63) & ~(size_t)63; return p; };
  float* Apad  = alloc(NP * NP);
  float* S     = alloc(TT * NN * BB);
  float* MEANB = alloc(NN * BB);
  float* STDB  = alloc(NN * BB);
  float* Wg0n  = alloc((size_t)NN * 130 * 128);
  float* Wc0n  = alloc((size_t)NN * 130 * 64);
  float* Wg1n  = alloc((size_t)NN * 256 * 128);
  float* Wc1n  = alloc((size_t)NN * 256 * 64);
  float* Bg0n  = alloc(NN * 128);
  float* Bc0n  = alloc(NN * 64);
  float* Bg1n  = alloc(NN * 128);
  float* Bc1n  = alloc(NN * 64);
  float* Rbuf  = alloc((size_t)NP * 128 * 64);
  float* XG1   = alloc((size_t)NP * 128 * 64);   // padded to 208 rows for guard-free stores
  float* ZR    = alloc((size_t)NN * 128 * 64);
  float* HC    = alloc((size_t)NN * 64 * 64);
  float* H     = alloc((size_t)NN * 64 * 64);
  float* CUR0  = alloc((size_t)TT * NN * 64 * 64);
  float* buf1  = alloc(NN * 8 * 91);
  float* buf2  = alloc(NN * 16 * 82);
  float* FX    = alloc(NN * ED);
  float* m1 = alloc(8);  float* r1 = alloc(8);
  float* m2 = alloc(16); float* r2 = alloc(16);
  float* mf = alloc(ED); float* rf = alloc(ED);
  float* ADJ = alloc(NN * NN);

  // ---- per-launch zero init (pad rows must be 0; deterministic across replays) ----
  k_zero<<<512, 256, 0, stream>>>(Apad, NP * NP);
  k_zero<<<1024, 256, 0, stream>>>(Rbuf, NP * 128 * 64);

  // ---- adjacency front-end ----
  k_conv1<<<cdiv(NN * 8 * 91, 256), 256, 0, stream>>>(NF, C1W, C1B, buf1);
  k_stats_conv<<<8, 256, 0, stream>>>(buf1, 8, 91, m1, r1);
  k_conv2<<<cdiv(NN * 16 * 82, 256), 256, 0, stream>>>(buf1, m1, r1, C2W, C2B, buf2);
  k_stats_conv<<<16, 256, 0, stream>>>(buf2, 16, 82, m2, r2);
  k_fc<<<cdiv(NN * ED, 256), 256, 0, stream>>>(buf2, m2, r2, FCW, FCB, FX);
  k_stats_feat<<<ED, 256, 0, stream>>>(FX, ED, mf, rf);
  k_pairs<<<cdiv(NN * NN, 256), 256, 0, stream>>>(FX, mf, rf, FOW, FOB, FC2W, FC2B, U, ADJ);
  k_arow<<<NN, 256, 0, stream>>>(E, ADJ, Apad);

  // ---- temporal normalization ----
  k_norm<<<cdiv(BB * NN, 256), 256, 0, stream>>>(Xe, S, MEANB, STDB);

  // ---- hoisted node-adapted weights ----
  k_wn<<<cdiv(NN * 130 * 128, 256), 256, 0, stream>>>(E, Wg[0], Wg0n, 130, 128, NN * 130 * 128);
  k_wn<<<cdiv(NN * 130 * 64, 256), 256, 0, stream>>>(E, Wc[0], Wc0n, 130, 64, NN * 130 * 64);
  k_wn<<<cdiv(NN * 256 * 128, 256), 256, 0, stream>>>(E, Wg[1], Wg1n, 256, 128, NN * 256 * 128);
  k_wn<<<cdiv(NN * 256 * 64, 256), 256, 0, stream>>>(E, Wc[1], Wc1n, 256, 64, NN * 256 * 64);
  k_bias<<<cdiv(NN * 128, 256), 256, 0, stream>>>(E, Bg[0], Bg0n, 128);
  k_bias<<<cdiv(NN * 64, 256), 256, 0, stream>>>(E, Bc[0], Bc0n, 64);
  k_bias<<<cdiv(NN * 128, 256), 256, 0, stream>>>(E, Bg[1], Bg1n, 128);
  k_bias<<<cdiv(NN * 64, 256), 256, 0, stream>>>(E, Bc[1], Bc1n, 64);

  const float* WgN[2] = {Wg0n, Wg1n};
  const float* BgN[2] = {Bg0n, Bg1n};
  const float* WcN[2] = {Wc0n, Wc1n};
  const float* BcN[2] = {Bc0n, Bc1n};

  // ---- 2-layer, 12-step recurrence ----
  for (int l = 0; l < 2; ++l) {
    k_zero<<<1024, 256, 0, stream>>>(H, NN * DH * BB);
    int dix = (l == 0) ? 1 : 64;
    int di = dix + DH;
    int KD = 2 * di;
    int KC = (KD + 15) / 16;
    int cols = di * 64;
    int buildN = NN * di * 64;
    for (int t = 0; t < TT; ++t) {
      const float* Xslice = (l == 0) ? (S + (size_t)t * NN * BB)
                                     : (CUR0 + (size_t)t * NN * DH * BB);
      // gate path
      k_build<<<cdiv(buildN, 256), 256, 0, stream>>>(Xslice, dix, H, nullptr, 0, Rbuf, di, buildN);
      k_supports<<<dim3(13, cols / 64), 128, 0, stream>>>(Apad, Rbuf, XG1, cols);
      k_gconv<0><<<dim3(NN, 8), 128, 0, stream>>>(Rbuf, XG1, WgN[l], BgN[l], ZR, di, 128, KC);
      // candidate path
      k_build<<<cdiv(buildN, 256), 256, 0, stream>>>(Xslice, dix, H, ZR, 1, Rbuf, di, buildN);
      k_supports<<<dim3(13, cols / 64), 128, 0, stream>>>(Apad, Rbuf, XG1, cols);
      k_gconv<1><<<dim3(NN, 4), 128, 0, stream>>>(Rbuf, XG1, WcN[l], BcN[l], HC, di, 64, KC);
      // state update (+ record layer-0 history)
      float* cur = (l == 0) ? (CUR0 + (size_t)t * NN * DH * BB) : nullptr;
      k_update<<<cdiv(NN * DH * BB, 256), 256, 0, stream>>>(ZR, HC, H, cur, (l == 0) ? 1 : 0);
    }
  }

  // ---- output head ----
  k_out<<<cdiv(BB * HH * NN, 256), 256, 0, stream>>>(H, ECW, ECB, EPS, STDB, MEANB, (float*)d_out);
}